// GNN_node_expander_29506425324086
// MI455X (gfx1250) — compile-verified
//
#include <hip/hip_runtime.h>

// GNN_node expander GIN forward for MI455X (gfx1250, wave32, WMMA f32 16x16x4)

#define DD 128   // embedding dim
#define HH 256   // hidden dim
#define LL 4     // layers
#define MT 80    // M-tile rows per block (5 x 16 sub-tiles); 100000 % 80 == 0
#define SUBT 5   // sub-tiles per wave
#define LDA (DD + 2)   // padded LDS row strides (stride mod 64 == 2 -> no bank conflicts)
#define LDU (HH + 2)

typedef __attribute__((ext_vector_type(2))) float v2f;
typedef __attribute__((ext_vector_type(8))) float v8f;

static __host__ __device__ inline int ceil_div_i(int a, int b) { return (a + b - 1) / b; }

// ---------------------------------------------------------------- init h
__global__ void k_init(const int* __restrict__ x, const int* __restrict__ mask,
                       const float* __restrict__ emb, float* __restrict__ h, int n) {
  int idx = blockIdx.x * blockDim.x + threadIdx.x;
  if (idx >= n) return;
  int i = idx >> 7;          // / DD
  int d = idx & (DD - 1);
  h[idx] = mask[i] ? emb[x[i] * DD + d] : 0.0f;
}

// ---------------------------------------------------------------- zero
__global__ void k_zero(float* __restrict__ p, int n) {
  int i = blockIdx.x * blockDim.x + threadIdx.x;
  if (i < n) p[i] = 0.0f;
}

// ---------------------------------------------------------------- message + scatter
// One wave per edge; lane handles 4 of 128 dims. msg = relu(h[src] (+ edge_attr@eW + eB))
template <bool HAS_EDGE>
__global__ void k_scatter(const float* __restrict__ h, const int* __restrict__ src,
                          const int* __restrict__ dst, int nE,
                          const float* __restrict__ eattr, const float* __restrict__ eW,
                          const float* __restrict__ eB, float* __restrict__ agg) {
  int gid = blockIdx.x * blockDim.x + threadIdx.x;
  int e = gid >> 5;
  if (e >= nE) return;
  int lane = gid & 31;
  int c = lane * 4;
  int s = src[e], d = dst[e];
  float4 hv = *reinterpret_cast<const float4*>(h + (size_t)s * DD + c);
  float m0 = hv.x, m1 = hv.y, m2 = hv.z, m3 = hv.w;
  if (HAS_EDGE) {
    float e0 = eB[c + 0], e1 = eB[c + 1], e2 = eB[c + 2], e3 = eB[c + 3];
#pragma unroll
    for (int j = 0; j < 7; ++j) {
      float a = eattr[e * 7 + j];
      const float4 wv = *reinterpret_cast<const float4*>(eW + j * DD + c);
      e0 = fmaf(a, wv.x, e0); e1 = fmaf(a, wv.y, e1);
      e2 = fmaf(a, wv.z, e2); e3 = fmaf(a, wv.w, e3);
    }
    m0 += e0; m1 += e1; m2 += e2; m3 += e3;
  }
  m0 = fmaxf(m0, 0.0f); m1 = fmaxf(m1, 0.0f);
  m2 = fmaxf(m2, 0.0f); m3 = fmaxf(m3, 0.0f);
  float* ap = agg + (size_t)d * DD + c;
  atomicAdd(ap + 0, m0); atomicAdd(ap + 1, m1);
  atomicAdd(ap + 2, m2); atomicAdd(ap + 3, m3);
}

// ---------------------------------------------------------------- GEMM1: t = z @ W1 + b1
// z = (1+eps)*h + agg fused into the LDS stage. Block: 80 rows, 8 waves; each wave does
// 5 M sub-tiles x one 16-col tile per pass (B fragment reused 5x). Fused BN1 column stats.
__global__ void k_gemm1(const float* __restrict__ h, const float* __restrict__ agg,
                        const float* __restrict__ epsPtr, const float* __restrict__ W1,
                        const float* __restrict__ b1, float* __restrict__ t,
                        float* __restrict__ sum1, float* __restrict__ sq1, int nRows) {
  __shared__ float ldsZ[MT * LDA];
  int mBase = blockIdx.x * MT;
  int tid = threadIdx.x;
  float epsv = 1.0f + epsPtr[0];
#pragma unroll
  for (int i = 0; i < (MT * DD) / 256; ++i) {   // 40 iters, coalesced
    int idx = i * 256 + tid;
    int row = idx >> 7;
    int col = idx & (DD - 1);
    int g = mBase + row;
    float z = 0.0f;
    if (g < nRows) z = fmaf(epsv, h[(size_t)g * DD + col], agg[(size_t)g * DD + col]);
    ldsZ[row * LDA + col] = z;
  }
  __syncthreads();
  int wave = tid >> 5, lane = tid & 31;
  int rsel = lane >> 4;          // lanes 16..31 carry K+2 (A/B) and M+8 (C)
  int rc = lane & 15;
  for (int n2 = 0; n2 < HH / 128; ++n2) {
    int col = (n2 * 8 + wave) * 16 + rc;
    float bias = b1[col];
    v8f acc[SUBT];
#pragma unroll
    for (int s2 = 0; s2 < SUBT; ++s2)
#pragma unroll
      for (int r = 0; r < 8; ++r) acc[s2][r] = bias;
    // partial unroll only: full unroll makes the scheduler preload A-frags and spill
#pragma unroll 2
    for (int ks = 0; ks < DD / 4; ++ks) {
      int kk = ks * 4 + rsel * 2;
      v2f b;
      b.x = W1[kk * HH + col];
      b.y = W1[(kk + 1) * HH + col];
#pragma unroll
      for (int s2 = 0; s2 < SUBT; ++s2) {
        v2f a;
        a.x = ldsZ[(s2 * 16 + rc) * LDA + kk];
        a.y = ldsZ[(s2 * 16 + rc) * LDA + kk + 1];
        acc[s2] = __builtin_amdgcn_wmma_f32_16x16x4_f32(false, a, false, b, (short)0,
                                                        acc[s2], false, false);
      }
    }
    // store + fused column stats (raw t, pre-BN)
    float s = 0.0f, ss = 0.0f;
#pragma unroll
    for (int s2 = 0; s2 < SUBT; ++s2) {
#pragma unroll
      for (int r = 0; r < 8; ++r) {
        int row = mBase + s2 * 16 + r + rsel * 8;
        if (row < nRows) {
          float v = acc[s2][r];
          t[(size_t)row * HH + col] = v;
          s += v;
          ss = fmaf(v, v, ss);
        }
      }
    }
    s += __shfl_xor(s, 16);
    ss += __shfl_xor(ss, 16);
    if (rsel == 0) {
      atomicAdd(&sum1[col], s);
      atomicAdd(&sq1[col], ss);
    }
  }
}

// ---------------------------------------------------------------- GEMM2: y = relu(BN1(t)) @ W2 + b2
// Fused BN2 column stats on raw y.
__global__ void k_gemm2(const float* __restrict__ t, const float* __restrict__ scale1,
                        const float* __restrict__ shift1, const float* __restrict__ W2,
                        const float* __restrict__ b2, float* __restrict__ y,
                        float* __restrict__ sum2, float* __restrict__ sq2, int nRows) {
  __shared__ float ldsU[MT * LDU];  // ~82.5 KB
  int mBase = blockIdx.x * MT;
  int tid = threadIdx.x;            // blockDim == 256 == HH
  float sc = scale1[tid], sh = shift1[tid];
#pragma unroll 4
  for (int i = 0; i < MT; ++i) {
    int g = mBase + i;
    float u = 0.0f;
    if (g < nRows) u = fmaxf(fmaf(t[(size_t)g * HH + tid], sc, sh), 0.0f);
    ldsU[i * LDU + tid] = u;
  }
  __syncthreads();
  int wave = tid >> 5, lane = tid & 31;
  int rsel = lane >> 4, rc = lane & 15;
  int col = wave * 16 + rc;         // DD == 128 == 8 waves * 16
  float bias = b2[col];
  v8f acc[SUBT];
#pragma unroll
  for (int s2 = 0; s2 < SUBT; ++s2)
#pragma unroll
    for (int r = 0; r < 8; ++r) acc[s2][r] = bias;
  // partial unroll only: avoid fragment preload spills to scratch
#pragma unroll 2
  for (int ks = 0; ks < HH / 4; ++ks) {
    int kk = ks * 4 + rsel * 2;
    v2f b;
    b.x = W2[kk * DD + col];
    b.y = W2[(kk + 1) * DD + col];
#pragma unroll
    for (int s2 = 0; s2 < SUBT; ++s2) {
      v2f a;
      a.x = ldsU[(s2 * 16 + rc) * LDU + kk];
      a.y = ldsU[(s2 * 16 + rc) * LDU + kk + 1];
      acc[s2] = __builtin_amdgcn_wmma_f32_16x16x4_f32(false, a, false, b, (short)0,
                                                      acc[s2], false, false);
    }
  }
  float s = 0.0f, ss = 0.0f;
#pragma unroll
  for (int s2 = 0; s2 < SUBT; ++s2) {
#pragma unroll
    for (int r = 0; r < 8; ++r) {
      int row = mBase + s2 * 16 + r + rsel * 8;
      if (row < nRows) {
        float v = acc[s2][r];
        y[(size_t)row * DD + col] = v;
        s += v;
        ss = fmaf(v, v, ss);
      }
    }
  }
  s += __shfl_xor(s, 16);
  ss += __shfl_xor(ss, 16);
  if (rsel == 0) {
    atomicAdd(&sum2[col], s);
    atomicAdd(&sq2[col], ss);
  }
}

// ---------------------------------------------------------------- BN finalize -> scale/shift
__global__ void k_bnfin(const float* __restrict__ sum, const float* __restrict__ sumsq,
                        const float* __restrict__ g, const float* __restrict__ b,
                        float invN, float* __restrict__ scale, float* __restrict__ shift,
                        int ncols) {
  int c = blockIdx.x * blockDim.x + threadIdx.x;
  if (c >= ncols) return;
  float mu = sum[c] * invN;
  float var = fmaf(-mu, mu, sumsq[c] * invN);
  float rstd = rsqrtf(var + 1e-5f);
  float sc = g[c] * rstd;
  scale[c] = sc;
  shift[c] = fmaf(-mu, sc, b[c]);
}

// ---------------------------------------------------------------- BN2 (+relu) + residual
__global__ void k_combine(const float* __restrict__ y, const float* __restrict__ scale,
                          const float* __restrict__ shift, const float* __restrict__ hin,
                          float* __restrict__ out, int n, int relu) {
  int idx = blockIdx.x * blockDim.x + threadIdx.x;
  if (idx >= n) return;
  int c = idx & (DD - 1);
  float v = fmaf(y[idx], scale[c], shift[c]);
  if (relu) v = fmaxf(v, 0.0f);
  out[idx] = v + hin[idx];
}

// ================================================================ host side
static void run_conv(hipStream_t stream, int N,
                     float* h, const int* src, const int* dst, int nE, bool hasEdge,
                     const float* eattr, const float* eW, const float* eB,
                     const float* epsPtr,
                     const float* W1, const float* b1, const float* g1, const float* bb1,
                     const float* W2, const float* b2, const float* bng, const float* bnb,
                     bool relu, float* outPtr,
                     float* agg, float* t, float* y, float* stats) {
  // stats layout: sum1[HH] sq1[HH] sum2[DD] sq2[DD] | scale1[HH] shift1[HH] scale2[DD] shift2[DD]
  float* sum1 = stats;
  float* sq1 = sum1 + HH;
  float* sum2 = sq1 + HH;
  float* sq2 = sum2 + DD;
  float* scale1 = sq2 + DD;
  float* shift1 = scale1 + HH;
  float* scale2 = shift1 + HH;
  float* shift2 = scale2 + DD;
  const int nd = N * DD;
  const float invN = 1.0f / (float)N;
  const int gemmBlocks = ceil_div_i(N, MT);

  k_zero<<<ceil_div_i(nd, 256), 256, 0, stream>>>(agg, nd);
  k_zero<<<ceil_div_i(2 * (HH + DD), 256), 256, 0, stream>>>(sum1, 2 * (HH + DD));
  if (hasEdge)
    k_scatter<true><<<ceil_div_i(nE * 32, 256), 256, 0, stream>>>(h, src, dst, nE,
                                                                  eattr, eW, eB, agg);
  else
    k_scatter<false><<<ceil_div_i(nE * 32, 256), 256, 0, stream>>>(h, src, dst, nE,
                                                                   nullptr, nullptr, nullptr, agg);
  k_gemm1<<<gemmBlocks, 256, 0, stream>>>(h, agg, epsPtr, W1, b1, t, sum1, sq1, N);
  k_bnfin<<<1, HH, 0, stream>>>(sum1, sq1, g1, bb1, invN, scale1, shift1, HH);
  k_gemm2<<<gemmBlocks, 256, 0, stream>>>(t, scale1, shift1, W2, b2, y, sum2, sq2, N);
  k_bnfin<<<1, DD, 0, stream>>>(sum2, sq2, bng, bnb, invN, scale2, shift2, DD);
  k_combine<<<ceil_div_i(nd, 256), 256, 0, stream>>>(y, scale2, shift2, h, outPtr, nd,
                                                     relu ? 1 : 0);
}

extern "C" void kernel_launch(void* const* d_in, const int* in_sizes, int n_in,
                              void* d_out, int out_size, void* d_ws, size_t ws_size,
                              hipStream_t stream) {
  const int*   x        = (const int*)d_in[0];
  const int*   eidx     = (const int*)d_in[1];
  const int*   xeidx    = (const int*)d_in[2];
  const int*   mask     = (const int*)d_in[3];
  const float* eattr    = (const float*)d_in[4];
  const float* node_emb = (const float*)d_in[5];
  const float* eps      = (const float*)d_in[6];
  const float* cW1      = (const float*)d_in[7];
  const float* cb1      = (const float*)d_in[8];
  const float* cg1      = (const float*)d_in[9];
  const float* cbb1     = (const float*)d_in[10];
  const float* cW2      = (const float*)d_in[11];
  const float* cb2      = (const float*)d_in[12];
  const float* eW       = (const float*)d_in[13];
  const float* eB       = (const float*)d_in[14];
  const float* bng      = (const float*)d_in[15];
  const float* bnb      = (const float*)d_in[16];
  const float* leps     = (const float*)d_in[17];
  const float* lW1      = (const float*)d_in[18];
  const float* lb1      = (const float*)d_in[19];
  const float* lg1      = (const float*)d_in[20];
  const float* lbb1     = (const float*)d_in[21];
  const float* lW2      = (const float*)d_in[22];
  const float* lb2      = (const float*)d_in[23];
  const float* lbng     = (const float*)d_in[24];
  const float* lbnb     = (const float*)d_in[25];
  const float* reps     = (const float*)d_in[26];
  const float* rW1      = (const float*)d_in[27];
  const float* rb1      = (const float*)d_in[28];
  const float* rg1      = (const float*)d_in[29];
  const float* rbb1     = (const float*)d_in[30];
  const float* rW2      = (const float*)d_in[31];
  const float* rb2      = (const float*)d_in[32];
  const float* rbng     = (const float*)d_in[33];
  const float* rbnb     = (const float*)d_in[34];

  const int N  = in_sizes[0];
  const int E  = in_sizes[1] / 2;
  const int EX = in_sizes[2] / 2;

  float* h    = (float*)d_ws;
  float* agg  = h + (size_t)N * DD;
  float* t    = agg + (size_t)N * DD;
  float* y    = t + (size_t)N * HH;
  float* stat = y + (size_t)N * DD;
  (void)ws_size; (void)n_in; (void)out_size;

  const int nd = N * DD;
  k_init<<<ceil_div_i(nd, 256), 256, 0, stream>>>(x, mask, node_emb, h, nd);

  const int* src = eidx;       const int* dst = eidx + E;
  const int* es  = xeidx;      const int* ed  = xeidx + EX;

  for (int l = 0; l < LL; ++l) {
    bool last = (l == LL - 1);
    // base GIN conv (+outer BN, relu unless last, residual)
    run_conv(stream, N, h, src, dst, E, /*hasEdge=*/true,
             eattr, eW + (size_t)l * 7 * DD, eB + (size_t)l * DD, eps + l,
             cW1 + (size_t)l * DD * HH, cb1 + (size_t)l * HH,
             cg1 + (size_t)l * HH, cbb1 + (size_t)l * HH,
             cW2 + (size_t)l * HH * DD, cb2 + (size_t)l * DD,
             bng + (size_t)l * DD, bnb + (size_t)l * DD,
             /*relu=*/!last, last ? (float*)d_out : h, agg, t, y, stat);
    if (!last) {
      // expander left: real -> expander (src=es, dst=ed)
      run_conv(stream, N, h, es, ed, EX, /*hasEdge=*/false,
               nullptr, nullptr, nullptr, leps + l,
               lW1 + (size_t)l * DD * HH, lb1 + (size_t)l * HH,
               lg1 + (size_t)l * HH, lbb1 + (size_t)l * HH,
               lW2 + (size_t)l * HH * DD, lb2 + (size_t)l * DD,
               lbng + (size_t)l * DD, lbnb + (size_t)l * DD,
               /*relu=*/true, h, agg, t, y, stat);
      // expander right: reversed edges (src=ed, dst=es)
      run_conv(stream, N, h, ed, es, EX, /*hasEdge=*/false,
               nullptr, nullptr, nullptr, reps + l,
               rW1 + (size_t)l * DD * HH, rb1 + (size_t)l * HH,
               rg1 + (size_t)l * HH, rbb1 + (size_t)l * HH,
               rW2 + (size_t)l * HH * DD, rb2 + (size_t)l * DD,
               rbng + (size_t)l * DD, rbnb + (size_t)l * DD,
               /*relu=*/true, h, agg, t, y, stat);
    }
  }
}